// ConvolutionBlock_82514911691300
// MI455X (gfx1250) — compile-verified
//
#include <hip/hip_runtime.h>
#include <math.h>

typedef __attribute__((ext_vector_type(2))) float v2f;
typedef __attribute__((ext_vector_type(8))) float v8f;

#define D3  110592     // 48*48*48 (unpadded channel stride)
#define DD  2304       // 48*48
#define ZP  52         // padded z extent (halo 2 each side)
#define ZP2 104        // ZP*2 floats per (x,y) row in pair-packed layout
#define PCH 239616     // 48*48*52*2 : pair-channel stride (floats)

__device__ __forceinline__ v8f wmma4(v2f a, v2f b, v8f c) {
  // D(16x16,f32) = A(16x4,f32) * B(4x16,f32) + C
  return __builtin_amdgcn_wmma_f32_16x16x4_f32(
      false, a, false, b, (short)0, c, false, false);
}

// ---------------------------------------------------------------------------
// Weight element values (math identical to reference kernel1/kernel2)
// ---------------------------------------------------------------------------
__device__ __forceinline__ float w1_elem(int t, int ci, int co,
                                         const float* __restrict__ w1_ss,
                                         const float* __restrict__ w1_vv0,
                                         const float* __restrict__ w1_vv1,
                                         const float (*emb)[5],
                                         const float (*sh1)[3]) {
  const float inv_nvox = 1.0f / 125.0f;
  if (ci < 4) {
    if (co >= 24) return 0.0f;
    float r = 0.0f;
    for (int b = 0; b < 5; ++b) r += emb[t][b] * w1_ss[b * 96 + ci * 24 + co];
    return 0.3535533905932738f * inv_nvox * r;  // PW0_1
  }
  const int u = (ci - 4) / 3, i = (ci - 4) % 3;
  if (co < 24) {
    float r = 0.0f;
    for (int b = 0; b < 5; ++b) r += emb[t][b] * w1_vv0[b * 96 + u * 24 + co];
    return 0.2041241452319315f * inv_nvox * r * sh1[t][i];  // PW0_1/sqrt3
  }
  const int w = (co - 24) / 3, kq = (co - 24) % 3;
  float r = 0.0f;
  for (int b = 0; b < 5; ++b) r += emb[t][b] * w1_vv1[b * 32 + u * 8 + w];
  const float s0 = sh1[t][0], s1 = sh1[t][1], s2 = sh1[t][2];
  float Mv;
  if (i == 0)      Mv = (kq == 1) ? -s2 : ((kq == 2) ? s1 : 0.0f);
  else if (i == 1) Mv = (kq == 0) ? s2 : ((kq == 2) ? -s0 : 0.0f);
  else             Mv = (kq == 0) ? -s1 : ((kq == 1) ? s0 : 0.0f);
  return 0.3535533905932738f * inv_nvox * r * Mv;  // sqrt(3/4)/sqrt6
}

__device__ __forceinline__ float w2_elem(int t, int ci, int co,
                                         const float* __restrict__ w2_ss,
                                         const float* __restrict__ w2_vs,
                                         const float (*emb)[5]) {
  const float inv_nvox = 1.0f / 125.0f;
  if (ci < 16 && co < 24) {
    float r = 0.0f;
    for (int b = 0; b < 5; ++b) r += emb[t][b] * w2_ss[b * 384 + ci * 24 + co];
    return 0.25f * inv_nvox * r;  // PW0_2
  }
  if (ci >= 16 && co >= 24) {
    const int u = (ci - 16) / 3, i = (ci - 16) % 3;
    const int w = (co - 24) / 3, kq = (co - 24) % 3;
    if (i == kq) {
      float r = 0.0f;
      for (int b = 0; b < 5; ++b) r += emb[t][b] * w2_vs[b * 64 + u * 8 + w];
      return 0.3535533905932738f * inv_nvox * r;  // sqrt(3/8)/sqrt3
    }
  }
  return 0.0f;
}

// ---------------------------------------------------------------------------
// Weight generation into WMMA B-fragment order:
//   Wp[t][kg][n][lane][2] ; lane = half*16+m ; element (row,col) =
//   (kg*4 + 2*half + par, n*16 + m)
// ---------------------------------------------------------------------------
__global__ __launch_bounds__(256) void build_weights(
    const float* __restrict__ w1_ss, const float* __restrict__ w1_vv0,
    const float* __restrict__ w1_vv1, const float* __restrict__ w2_ss,
    const float* __restrict__ w2_vs, float* __restrict__ W1p,
    float* __restrict__ W2p) {
  __shared__ float emb[125][5];
  __shared__ float sh1[125][3];
  const int tid = threadIdx.x;
  if (tid < 125) {
    const int i = tid / 25, j = (tid / 5) % 5, k = tid % 5;
    const float a0 = (float)(i - 2), a1 = (float)(j - 2), a2 = (float)(k - 2);
    const float norm = sqrtf(a0 * a0 + a1 * a1 + a2 * a2);
    const float step = 2.5f / 6.0f;
    const float cemb = 1.14136f * expf(2.0f);
    for (int b = 0; b < 5; ++b) {
      const float val = step * (float)(b + 1);
      const float diff = (norm - val) / step;
      const float t1 = diff + 1.0f, t2 = 1.0f - diff;
      const float u1 = (t1 > 0.0f) ? expf(-1.0f / t1) : 0.0f;
      const float u2 = (t2 > 0.0f) ? expf(-1.0f / t2) : 0.0f;
      emb[tid][b] = cemb * u1 * u2;
    }
    const float inv = (norm > 0.0f) ? 1.7320508075688772f / norm : 0.0f;
    sh1[tid][0] = a0 * inv; sh1[tid][1] = a1 * inv; sh1[tid][2] = a2 * inv;
  }
  __syncthreads();

  // W1p: 125 * 4 * 3 * 64
  for (int idx = blockIdx.x * blockDim.x + tid; idx < 125 * 4 * 192;
       idx += gridDim.x * blockDim.x) {
    const int t = idx / 768;
    const int rem = idx % 768;
    const int kg = rem / 192;
    const int r2 = rem % 192;
    const int n = r2 / 64, le = r2 % 64;
    const int lane = le >> 1, par = le & 1;
    const int half = lane >> 4, m = lane & 15;
    const int ci = kg * 4 + 2 * half + par;
    const int co = n * 16 + m;
    W1p[idx] = w1_elem(t, ci, co, w1_ss, w1_vv0, w1_vv1, emb, sh1);
  }
  // W2p: 125 * 10 * 3 * 64
  for (int idx = blockIdx.x * blockDim.x + tid; idx < 125 * 10 * 192;
       idx += gridDim.x * blockDim.x) {
    const int t = idx / 1920;
    const int rem = idx % 1920;
    const int kg = rem / 192;
    const int r2 = rem % 192;
    const int n = r2 / 64, le = r2 % 64;
    const int lane = le >> 1, par = le & 1;
    const int half = lane >> 4, m = lane & 15;
    const int ci = kg * 4 + 2 * half + par;
    const int co = n * 16 + m;
    W2p[idx] = w2_elem(t, ci, co, w2_ss, w2_vs, emb);
  }
}

// C_SIG = 1/sqrt( sum(sigmoid(z)^2 * pdf(z)) * dz ), z = linspace(-12,12,100001)
__global__ __launch_bounds__(256) void csig_kernel(float* __restrict__ out) {
  __shared__ double red[256];
  double s = 0.0;
  const double dz = 24.0 / 100000.0;
  for (int i = threadIdx.x; i < 100001; i += 256) {
    const double z = -12.0 + dz * (double)i;
    const double pdf = exp(-0.5 * z * z) * 0.3989422804014327;
    const double sg = 1.0 / (1.0 + exp(-z));
    s += sg * sg * pdf;
  }
  red[threadIdx.x] = s;
  __syncthreads();
  for (int o = 128; o > 0; o >>= 1) {
    if (threadIdx.x < o) red[threadIdx.x] += red[threadIdx.x + o];
    __syncthreads();
  }
  if (threadIdx.x == 0) out[0] = (float)(1.0 / sqrt(red[0] * dz));
}

// ---------------------------------------------------------------------------
// Pack x (B,16,48,48,48) -> pair layout [b][8][x][y][ZP][2] with z halo zeros
// ---------------------------------------------------------------------------
__global__ __launch_bounds__(256) void pack_x_kernel(
    const float* __restrict__ x, float* __restrict__ xp) {
  const int idx = blockIdx.x * blockDim.x + threadIdx.x;  // b,pair,x,y,zp
  const int total = 4 * 8 * DD * ZP;
  if (idx >= total) return;
  const int zp = idx % ZP;
  int rest = idx / ZP;
  const int y = rest % 48; rest /= 48;
  const int xx = rest % 48; rest /= 48;
  const int pair = rest % 8;
  const int b = rest / 8;
  const int z = zp - 2;
  float v0 = 0.0f, v1 = 0.0f;
  if ((unsigned)z < 48u) {
    const float* s = x + (b * 16 + pair * 2) * D3 + xx * DD + y * 48 + z;
    v0 = s[0];
    v1 = s[D3];
  }
  xp[idx * 2] = v0;
  xp[idx * 2 + 1] = v1;
}

// Zero the z-halo of the packed h tensor [b][20][x][y][ZP][2]
__global__ __launch_bounds__(256) void zero_h_borders(float* __restrict__ hp) {
  const int idx = blockIdx.x * blockDim.x + threadIdx.x;  // b,pair,x,y
  const int total = 4 * 20 * DD;
  if (idx >= total) return;
  float* p = hp + idx * ZP2;
#pragma unroll
  for (int i = 0; i < 4; ++i) { p[i] = 0.0f; p[100 + i] = 0.0f; }
}

// ---------------------------------------------------------------------------
// WMMA conv core, M=48 (whole z column): acc[mt][n], mt z-tile, n outch-tile.
// Per k-group: 3 A-b64 + 3 B-b64 -> 9 WMMAs.  xx/yy/b are wave-uniform
// scalars (readfirstlane'd by caller) so bounds checks are SALU branches.
// ---------------------------------------------------------------------------
template <int KG>
__device__ __forceinline__ void conv_accum48(const float* __restrict__ inpb,
                                             const float* __restrict__ Wp,
                                             int xx, int yy, int m, int half,
                                             int lane, v8f acc[3][3]) {
  for (int dx = 0; dx < 5; ++dx) {
    const int xs = xx + dx - 2;
    if ((unsigned)xs >= 48u) continue;
    for (int dy = 0; dy < 5; ++dy) {
      const int ys = yy + dy - 2;
      if ((unsigned)ys >= 48u) continue;
      const float* irow = inpb + (xs * 48 + ys) * ZP2;
      const float* wtap = Wp + ((dx * 5 + dy) * 5) * (KG * 192) + lane * 2;
#pragma unroll
      for (int dz = 0; dz < 5; ++dz) {
        const float* wt = wtap + dz * (KG * 192);
        const int zb2 = (m + dz) * 2;
        for (int kg = 0; kg < KG; ++kg) {
          const float* arow = irow + (kg * 2 + half) * PCH + zb2;
          const v2f a0 = *(const v2f*)(arow);
          const v2f a1 = *(const v2f*)(arow + 32);   // +16 z
          const v2f a2 = *(const v2f*)(arow + 64);   // +32 z
          const float* wr = wt + kg * 192;
          const v2f b0 = *(const v2f*)(wr);
          const v2f b1 = *(const v2f*)(wr + 64);
          const v2f b2 = *(const v2f*)(wr + 128);
          acc[0][0] = wmma4(a0, b0, acc[0][0]);
          acc[0][1] = wmma4(a0, b1, acc[0][1]);
          acc[0][2] = wmma4(a0, b2, acc[0][2]);
          acc[1][0] = wmma4(a1, b0, acc[1][0]);
          acc[1][1] = wmma4(a1, b1, acc[1][1]);
          acc[1][2] = wmma4(a1, b2, acc[1][2]);
          acc[2][0] = wmma4(a2, b0, acc[2][0]);
          acc[2][1] = wmma4(a2, b1, acc[2][1]);
          acc[2][2] = wmma4(a2, b2, acc[2][2]);
        }
      }
    }
  }
}

// ---------------------------------------------------------------------------
// Conv1 (Cin=16) + skip + gate -> packed h [b][20][x][y][ZP][2]
// One wave per (b, x, y); covers all 48 z and all 48 out channels.
// ---------------------------------------------------------------------------
__global__ __launch_bounds__(256) void conv1_kernel(
    const float* __restrict__ x, const float* __restrict__ xpad,
    const float* __restrict__ W1p, const float* __restrict__ sc1_w,
    const float* __restrict__ csig, float* __restrict__ hpad) {
  const int lane = threadIdx.x & 31;
  const int wid = __builtin_amdgcn_readfirstlane(
      blockIdx.x * (blockDim.x >> 5) + (threadIdx.x >> 5));
  const int yy = wid % 48;
  const int xx = (wid / 48) % 48;
  const int b = wid / 2304;
  if (b >= 4) return;
  const int m = lane & 15;
  const int half = lane >> 4;

  v8f acc[3][3] = {};
  conv_accum48<4>(xpad + b * 8 * PCH, W1p, xx, yy, m, half, lane, acc);

  // fused skip + gate + store into packed h
  const float CS = *csig;
  const float* xpos = x + b * 16 * D3 + xx * DD + yy * 48;
  float* hpb = hpad + b * 20 * PCH + (xx * 48 + yy) * ZP2;
  const int baseL = half * 16;
  const int ch1s = 16 + (m & 7);
  const int u1 = ((m >= 8) ? (m - 8) : 0) / 3;
  const int u2 = (8 + m) / 3;
#pragma unroll
  for (int mt = 0; mt < 3; ++mt) {
#pragma unroll
    for (int r = 0; r < 8; ++r) {
      const int z = mt * 16 + r + half * 8;
      const int zo2 = (z + 2) * 2;
      float sk0 = 0.0f, sk1 = 0.0f;
#pragma unroll
      for (int u = 0; u < 4; ++u) {
        const float xv = xpos[u * D3 + z];
        sk0 += xv * sc1_w[u * 24 + m];
        sk1 += xv * sc1_w[u * 24 + ch1s];
      }
      const float y0 = acc[mt][0][r] + 0.5f * sk0;
      const float y1 = acc[mt][1][r] + ((m < 8) ? 0.5f * sk1 : 0.0f);
      const float y2 = acc[mt][2][r];
      hpb[(m >> 1) * PCH + zo2 + (m & 1)] =
          1.4142135623730951f * fmaxf(y0, 0.0f);
      const float gv = CS / (1.0f + __expf(-y1));  // valid on lanes m<8
      const float gA = __shfl(gv, baseL + u1, 32);
      const float gB = __shfl(gv, baseL + u2, 32);
      if (m >= 8) {
        const int ch = m + 8;  // 16 + (m-8)
        hpb[(ch >> 1) * PCH + zo2 + (ch & 1)] = y1 * gA;
      }
      const int ch2 = 24 + m;
      hpb[(ch2 >> 1) * PCH + zo2 + (ch2 & 1)] = y2 * gB;
    }
  }
}

// ---------------------------------------------------------------------------
// Conv2 (Cin=40) + skip2 + gate -> d_out (B,40,48,48,48)
// ---------------------------------------------------------------------------
__global__ __launch_bounds__(256) void conv2_kernel(
    const float* __restrict__ hpad, const float* __restrict__ W2p,
    const float* __restrict__ sc2_s, const float* __restrict__ sc2_v,
    const float* __restrict__ csig, float* __restrict__ out) {
  const int lane = threadIdx.x & 31;
  const int wid = __builtin_amdgcn_readfirstlane(
      blockIdx.x * (blockDim.x >> 5) + (threadIdx.x >> 5));
  const int yy = wid % 48;
  const int xx = (wid / 48) % 48;
  const int b = wid / 2304;
  if (b >= 4) return;
  const int m = lane & 15;
  const int half = lane >> 4;

  v8f acc[3][3] = {};
  conv_accum48<10>(hpad + b * 20 * PCH, W2p, xx, yy, m, half, lane, acc);

  const float CS = *csig;
  const float* hpb = hpad + b * 20 * PCH + (xx * 48 + yy) * ZP2;
  float* ob = out + b * 40 * D3 + xx * DD + yy * 48;
  const int baseL = half * 16;
  const int ch1s = 16 + (m & 7);
  const int vi1 = (m >= 8) ? (m - 8) : 0;
  const int w1i = vi1 / 3, i1 = vi1 % 3;
  const int vi2 = 8 + m;
  const int w2i = vi2 / 3, i2 = vi2 % 3;
#pragma unroll
  for (int mt = 0; mt < 3; ++mt) {
#pragma unroll
    for (int r = 0; r < 8; ++r) {
      const int z = mt * 16 + r + half * 8;
      const int zo2 = (z + 2) * 2;
      // scalar skips (pre-act channels < 24)
      float sk0 = 0.0f, sk1s = 0.0f;
#pragma unroll
      for (int u = 0; u < 16; ++u) {
        const float hv = hpb[(u >> 1) * PCH + zo2 + (u & 1)];
        sk0 += hv * sc2_s[u * 24 + m];
        sk1s += hv * sc2_s[u * 24 + ch1s];
      }
      // vector skips (pre-act channels >= 24)
      float sk1v = 0.0f, sk2v = 0.0f;
#pragma unroll
      for (int u = 0; u < 8; ++u) {
        const int c1 = 16 + u * 3 + i1;
        const int c2 = 16 + u * 3 + i2;
        sk1v += hpb[(c1 >> 1) * PCH + zo2 + (c1 & 1)] * sc2_v[u * 8 + w1i];
        sk2v += hpb[(c2 >> 1) * PCH + zo2 + (c2 & 1)] * sc2_v[u * 8 + w2i];
      }
      const float y0 = acc[mt][0][r] + 0.25f * sk0;
      const float y1 = acc[mt][1][r] +
                       ((m < 8) ? 0.25f * sk1s : 0.3535533905932738f * sk1v);
      const float y2 = acc[mt][2][r] + 0.3535533905932738f * sk2v;
      ob[m * D3 + z] = 1.4142135623730951f * fmaxf(y0, 0.0f);
      const float gv = CS / (1.0f + __expf(-y1));
      const float gA = __shfl(gv, baseL + vi1 / 3, 32);
      const float gB = __shfl(gv, baseL + vi2 / 3, 32);
      if (m >= 8) ob[(16 + (m - 8)) * D3 + z] = y1 * gA;
      ob[(24 + m) * D3 + z] = y2 * gB;
    }
  }
}

// ---------------------------------------------------------------------------
extern "C" void kernel_launch(void* const* d_in, const int* in_sizes, int n_in,
                              void* d_out, int out_size, void* d_ws,
                              size_t ws_size, hipStream_t stream) {
  const float* x      = (const float*)d_in[0];
  const float* w1_ss  = (const float*)d_in[1];
  const float* w1_vv0 = (const float*)d_in[2];
  const float* w1_vv1 = (const float*)d_in[3];
  const float* w2_ss  = (const float*)d_in[4];
  const float* w2_vs  = (const float*)d_in[5];
  const float* sc1_w  = (const float*)d_in[6];
  const float* sc2_s  = (const float*)d_in[7];
  const float* sc2_v  = (const float*)d_in[8];

  float* ws = (float*)d_ws;
  float* W1p  = ws;                       // 96000 floats
  float* W2p  = ws + 96000;               // 240000 floats
  float* csig = ws + 336000;              // 1 float (+pad)
  float* xpad = ws + 336064;              // 4*8*PCH  = 7,667,712 floats
  float* hpad = ws + 336064 + 4 * 8 * PCH;  // 4*20*PCH = 19,169,280 floats

  build_weights<<<64, 256, 0, stream>>>(w1_ss, w1_vv0, w1_vv1, w2_ss, w2_vs,
                                        W1p, W2p);
  csig_kernel<<<1, 256, 0, stream>>>(csig);
  pack_x_kernel<<<(4 * 8 * DD * ZP + 255) / 256, 256, 0, stream>>>(x, xpad);
  zero_h_borders<<<(4 * 20 * DD + 255) / 256, 256, 0, stream>>>(hpad);

  const int blocks = (4 * 48 * 48) / 8;  // one wave per (b,x,y), full z column
  conv1_kernel<<<blocks, 256, 0, stream>>>(x, xpad, W1p, sc1_w, csig, hpad);
  conv2_kernel<<<blocks, 256, 0, stream>>>(hpad, W2p, sc2_s, sc2_v, csig,
                                           (float*)d_out);
}